// LDSMIXTURELQR_58445914964044
// MI455X (gfx1250) — compile-verified
//
#include <hip/hip_runtime.h>
#include <stdint.h>

// ---------------------------------------------------------------------------
// LDS-mixture-LQR sequential scan for MI455X (gfx1250).
// Serial nonlinear recurrence -> single wave32; CDNA5 async LDS DMA hides all
// global traffic (ASYNCcnt double-buffered input prefetch + async output
// stores from LDS staging). Inner loop tuned for minimum dependency-chain
// length: v_exp_f32 / v_rcp_f32 / v_sqrt_f32 + FMAs only.
// ---------------------------------------------------------------------------

#define T_STEPS 131072
#define CHUNK   256
#define NCHUNK  (T_STEPS / CHUNK)   // 512 chunks

// Low 32 bits of a generic pointer to a __shared__ object == LDS byte address
// (ISA 10.2: LDS aperture mapping truncates to addr[31:0]).
__device__ __forceinline__ unsigned lds_addr32(const void* p) {
  return (unsigned)(uintptr_t)p;
}

// ASYNC DMA: global -> LDS, 16B per lane per issue (tracked by ASYNCcnt).
__device__ __forceinline__ void async_ld_b128(unsigned lds, const void* gaddr) {
  asm volatile("global_load_async_to_lds_b128 %0, %1, off"
               :: "v"(lds), "v"(gaddr) : "memory");
}
// ASYNC DMA: LDS -> global, 16B per lane per issue.
__device__ __forceinline__ void async_st_b128(void* gaddr, unsigned lds) {
  asm volatile("global_store_async_from_lds_b128 %0, %1, off"
               :: "v"(gaddr), "v"(lds) : "memory");
}
// ASYNC DMA: LDS -> global, 4B per lane (for 4B-aligned-only output regions).
__device__ __forceinline__ void async_st_b32(void* gaddr, unsigned lds) {
  asm volatile("global_store_async_from_lds_b32 %0, %1, off"
               :: "v"(gaddr), "v"(lds) : "memory");
}

#define WAIT_ASYNC_LE8() asm volatile("s_wait_asynccnt 0x8" ::: "memory")
#define WAIT_ASYNC_0()   asm volatile("s_wait_asynccnt 0x0" ::: "memory")
#define WAIT_DS_0()      asm volatile("s_wait_dscnt 0x0"    ::: "memory")

#define LOG2E 1.4426950408889634f

__global__ __launch_bounds__(32)
void lds_mixture_lqr_scan(const float* __restrict__ xstar1,   // (T,2)
                          const float* __restrict__ xstar2,   // (T,2)
                          const float* __restrict__ init_y,   // (2,)
                          const float* __restrict__ x0p,      // (1,)
                          const float* __restrict__ Ap,       // (1,)
                          const float* __restrict__ Bp,       // (2,)
                          const float* __restrict__ g1p,      // (1,)
                          const float* __restrict__ g2p,      // (1,)
                          float* __restrict__ out) {
  // LDS staging: double-buffered inputs, single-buffered outputs.
  __shared__ __align__(16) float sIn1[2][CHUNK * 2];  // 2 x 2KB
  __shared__ __align__(16) float sIn2[2][CHUNK * 2];  // 2 x 2KB
  __shared__ __align__(16) float sX[CHUNK];           // 1KB  (pre-step x)
  __shared__ __align__(16) float sY[CHUNK * 2];       // 2KB  (pre-step y)
  __shared__ __align__(16) float sU[CHUNK * 2];       // 2KB  (u_t)

  const int lane = threadIdx.x;

  float* outX = out;                       // x_full: rows 0..T
  float* outY = out + (T_STEPS + 1);       // y_full: rows 0..T (2 floats each)
  float* outU = out + 3 * (T_STEPS + 1);   // us:     rows 0..T-1

  // Scalar parameters (tiny one-time loads).
  const float Aa   = Ap[0];
  const float Bsum = Bp[0] + Bp[1];        // dot(B, [n1,n1]) == (B0+B1)*n1
  const float ng1  = -g1p[0];
  const float ng2  = -g2p[0];
  const float kDT  = 1.0f / 60.0f;

  float x  = x0p[0];
  float y0 = init_y[0];
  float y1 = init_y[1];

  // ---- Prologue: async-prefetch chunk 0 inputs (8 ops = 4KB) ----
  {
    const char* p1 = (const char*)xstar1 + lane * 16;
    const char* p2 = (const char*)xstar2 + lane * 16;
    unsigned a1 = lds_addr32(&sIn1[0][0]) + lane * 16;
    unsigned a2 = lds_addr32(&sIn2[0][0]) + lane * 16;
#pragma unroll
    for (int s = 0; s < 4; ++s) {
      async_ld_b128(a1 + s * 512, p1 + s * 512);
      async_ld_b128(a2 + s * 512, p2 + s * 512);
    }
  }

  for (int c = 0; c < NCHUNK; ++c) {
    const int buf = c & 1;

    // Issue prefetch for chunk c+1, then wait so that chunk c's loads (and
    // chunk c-1's output stores) are complete: loads retire in order, so
    // ASYNCcnt <= 8 leaves only the 8 just-issued prefetch ops in flight.
    if (c + 1 < NCHUNK) {
      const char* p1 = (const char*)(xstar1 + (size_t)(c + 1) * CHUNK * 2) + lane * 16;
      const char* p2 = (const char*)(xstar2 + (size_t)(c + 1) * CHUNK * 2) + lane * 16;
      unsigned a1 = lds_addr32(&sIn1[buf ^ 1][0]) + lane * 16;
      unsigned a2 = lds_addr32(&sIn2[buf ^ 1][0]) + lane * 16;
#pragma unroll
      for (int s = 0; s < 4; ++s) {
        async_ld_b128(a1 + s * 512, p1 + s * 512);
        async_ld_b128(a2 + s * 512, p2 + s * 512);
      }
      WAIT_ASYNC_LE8();
    } else {
      WAIT_ASYNC_0();
    }

    // ---- Serial scan over this chunk (lane 0 only; the chain is scalar) ----
    if (lane == 0) {
      const float* __restrict__ in1 = sIn1[buf];
      const float* __restrict__ in2 = sIn2[buf];
#pragma unroll 4
      for (int i = 0; i < CHUNK; ++i) {
        const float xs1x = in1[2 * i], xs1y = in1[2 * i + 1];
        const float xs2x = in2[2 * i], xs2y = in2[2 * i + 1];

        // Stage pre-step state (row t = c*CHUNK + i); row 0 == x0/init_y.
        sX[i]         = x;
        sY[2 * i]     = y0;
        sY[2 * i + 1] = y1;

        // w = sigmoid(clip(x,-10,10)) = rcp(1 + exp2(-x*log2e))
        // v_max + v_min + v_mul + v_exp + v_add + v_rcp: all 1-instr forms.
        const float xc = fminf(fmaxf(x, -10.0f), 10.0f);
        const float ex = __builtin_amdgcn_exp2f(xc * (-LOG2E));
        const float w  = __builtin_amdgcn_rcpf(1.0f + ex);
        const float wm = 1.0f - w;

        const float e1x = y0 - xs1x, e1y = y1 - xs1y;
        const float e2x = y0 - xs2x, e2y = y1 - xs2y;

        const float ux = fmaf(ng1 * e1x, w, (ng2 * e2x) * wm);
        const float uy = fmaf(ng1 * e1y, w, (ng2 * e2y) * wm);
        sU[2 * i]     = ux;
        sU[2 * i + 1] = uy;

        // n1 = ||e1||: v_mul + v_fma + v_sqrt (single-instruction sqrt).
        const float n1 = __builtin_amdgcn_sqrtf(fmaf(e1x, e1x, e1y * e1y));
        x  = fmaf(Aa, x, Bsum * n1);   // LDS state update
        y0 = fmaf(kDT, ux, y0);        // A' == 1, B' == 1/60
        y1 = fmaf(kDT, uy, y1);
      }
    }

    // Fence LDS staging stores, then DMA them out (34 async ops; they drain
    // during the next chunk's prefetch window, never blocking compute).
    WAIT_DS_0();
    {
      // x region: base byte offset 1024*c -> 16B aligned -> b128 (2 ops).
      char*    gx = (char*)(outX + (size_t)c * CHUNK) + lane * 16;
      unsigned ax = lds_addr32(sX) + lane * 16;
#pragma unroll
      for (int s = 0; s < 2; ++s) async_st_b128(gx + s * 512, ax + s * 512);

      // y region starts at (T+1)*4 bytes == 4 (mod 16) -> only 4B aligned.
      char*    gy = (char*)(outY + (size_t)c * CHUNK * 2) + lane * 4;
      unsigned ay = lds_addr32(sY) + lane * 4;
#pragma unroll
      for (int s = 0; s < 16; ++s) async_st_b32(gy + s * 128, ay + s * 128);

      // u region starts at 3*(T+1)*4 bytes == 12 (mod 16) -> b32 as well.
      char*    gu = (char*)(outU + (size_t)c * CHUNK * 2) + lane * 4;
      unsigned au = lds_addr32(sU) + lane * 4;
#pragma unroll
      for (int s = 0; s < 16; ++s) async_st_b32(gu + s * 128, au + s * 128);
    }
  }

  // Drain all async stores, then write the final row T.
  WAIT_ASYNC_0();
  if (lane == 0) {
    outX[T_STEPS]         = x;
    outY[2 * T_STEPS]     = y0;
    outY[2 * T_STEPS + 1] = y1;
  }
}

extern "C" void kernel_launch(void* const* d_in, const int* in_sizes, int n_in,
                              void* d_out, int out_size, void* d_ws, size_t ws_size,
                              hipStream_t stream) {
  (void)in_sizes; (void)n_in; (void)out_size; (void)d_ws; (void)ws_size;
  const float* xstar1 = (const float*)d_in[0];
  const float* xstar2 = (const float*)d_in[1];
  const float* init_y = (const float*)d_in[2];
  const float* x0     = (const float*)d_in[3];
  const float* A      = (const float*)d_in[4];
  const float* B      = (const float*)d_in[5];
  const float* gain1  = (const float*)d_in[6];
  const float* gain2  = (const float*)d_in[7];
  float* out = (float*)d_out;

  // One wave32: the recurrence is a single serial dependency chain.
  lds_mixture_lqr_scan<<<dim3(1), dim3(32), 0, stream>>>(
      xstar1, xstar2, init_y, x0, A, B, gain1, gain2, out);
}